// BLOutputLayer_89069031785172
// MI455X (gfx1250) — compile-verified
//
#include <hip/hip_runtime.h>

// Scatter-add: out[idx[r], :] += in[r, :], out zero-initialized.
// N rows = 32768, C = 512 f32, slots = B*L = 65536.
// Pure bandwidth problem: ~192 MiB HBM traffic -> ~8.6 us floor at 23.3 TB/s.
// Output (128 MiB) fits in the 192 MB L2, so zero-fill + atomic double-write
// is absorbed by L2; HBM sees one writeback.

typedef float v4f __attribute__((ext_vector_type(4)));

#define CDIM 512           // features per row (reference C)
#define ROWS_PER_BLOCK 8   // 256 threads / wave32 = 8 waves, one row per wave

// Pass 1: zero the dense output with wide b128 stores (regular temporal hint so
// the lines are L2-resident when the atomics hit them).
__global__ __launch_bounds__(256) void scn_zero_kernel(float* __restrict__ out,
                                                       long long n4) {
  long long i = (long long)blockIdx.x * blockDim.x + threadIdx.x;
  const long long stride = (long long)gridDim.x * blockDim.x;
  v4f z = {0.0f, 0.0f, 0.0f, 0.0f};
  v4f* __restrict__ o = (v4f*)out;
  for (; i < n4; i += stride) {
    o[i] = z;
  }
}

// Pass 2: one wave32 per row. Each lane reads 4x float4 (coalesced 512B per
// chunk, non-temporal so the streamed input doesn't evict the L2-resident
// output), then does 16 native f32 atomic adds (no-return form, device scope).
__global__ __launch_bounds__(256) void scn_scatter_kernel(
    const float* __restrict__ in, const int* __restrict__ idx,
    float* __restrict__ out, int nrows) {
  const int lane = threadIdx.x & 31;          // wave32
  const int wave = threadIdx.x >> 5;
  const int row = blockIdx.x * ROWS_PER_BLOCK + wave;
  if (row >= nrows) return;

  const long long slot = (long long)idx[row];
  const v4f* __restrict__ src = (const v4f*)(in + (long long)row * CDIM);
  float* dst = out + slot * CDIM;

#pragma unroll
  for (int c = 0; c < 4; ++c) {
    // chunk c covers floats [c*128, c*128+128); lane handles 4 consecutive
    v4f v = __builtin_nontemporal_load(&src[c * 32 + lane]);
    float* p = dst + c * 128 + lane * 4;
    // Hardware L2 fadd, no return value (STOREcnt path), device scope.
    asm volatile("global_atomic_add_f32 %0, %1, off scope:SCOPE_DEV"
                 :: "v"(p), "v"(v.x) : "memory");
    asm volatile("global_atomic_add_f32 %0, %1, off offset:4 scope:SCOPE_DEV"
                 :: "v"(p), "v"(v.y) : "memory");
    asm volatile("global_atomic_add_f32 %0, %1, off offset:8 scope:SCOPE_DEV"
                 :: "v"(p), "v"(v.z) : "memory");
    asm volatile("global_atomic_add_f32 %0, %1, off offset:12 scope:SCOPE_DEV"
                 :: "v"(p), "v"(v.w) : "memory");
  }
}

extern "C" void kernel_launch(void* const* d_in, const int* in_sizes, int n_in,
                              void* d_out, int out_size, void* d_ws, size_t ws_size,
                              hipStream_t stream) {
  const float* in = (const float*)d_in[0];   // (N_ACTIVE, 512) f32
  const int* idx = (const int*)d_in[1];      // (N_ACTIVE,) int
  float* out = (float*)d_out;                // (B*L, 512) f32

  const int nrows = in_sizes[1];             // N_ACTIVE
  const long long n4 = (long long)out_size / 4;  // number of float4s in out

  // Zero-fill: 8M float4 stores spread over 8192 blocks (grid-stride).
  scn_zero_kernel<<<8192, 256, 0, stream>>>(out, n4);

  // Scatter: one wave per row, 8 rows per 256-thread block.
  const int sblocks = (nrows + ROWS_PER_BLOCK - 1) / ROWS_PER_BLOCK;
  scn_scatter_kernel<<<sblocks, 256, 0, stream>>>(in, idx, out, nrows);
}